// VectorQuantizer_23673859736374
// MI455X (gfx1250) — compile-verified
//
#include <hip/hip_runtime.h>
#include <hip/hip_bf16.h>

typedef __attribute__((ext_vector_type(2))) float v2f;
typedef __attribute__((ext_vector_type(4))) float v4f;
typedef __attribute__((ext_vector_type(8))) float v8f;
typedef __attribute__((ext_vector_type(4))) unsigned int v4u;
typedef __attribute__((ext_vector_type(4))) int v4i;
typedef __attribute__((ext_vector_type(8))) int v8i;

#define DIM   64
#define NEMB  512
#define NTOT  131072            // 32*64*64 rows
#define ESTR  68                // padded LDS row stride (floats): banks 4n+c -> conflict-free

// flat float offsets into d_out (tuple outputs concatenated in return order)
static const long OFF_DIST  = 0L;
static const long OFF_IDX   = 67108864L;            // 131072*512
static const long OFF_ENC   = 67239936L;            // + 131072
static const long OFF_QOUT  = 134348800L;           // + 131072*512
static const long OFF_QLOSS = 142737408L;           // + 8388608
static const long OFF_PERP  = 151126016L;           // + 8388608

__global__ void vq_zero_hist(unsigned int* hist) { hist[threadIdx.x] = 0u; }

__global__ __launch_bounds__(256) void vq_main(const float* __restrict__ in,
                                               const float* __restrict__ emb,
                                               float* __restrict__ out,
                                               unsigned int* __restrict__ hist)
{
    __shared__ float lds_emb[NEMB * ESTR];   // 139264 B, TDM-padded codebook
    __shared__ float lds_enorm[NEMB];
    __shared__ float lds_xnorm[8 * 16];
    __shared__ int   lds_argidx[8 * 16];

    const int tid  = threadIdx.x;
    const int lane = tid & 31;
    const int wave = tid >> 5;
    const int m    = lane & 15;     // row-in-strip / col-in-tile
    const int hi   = lane >> 4;     // wave half
    const int koff = hi * 2;        // K-offset inside a 4-wide WMMA step

    // ---- TDM: DMA the whole codebook (512x64 f32) into LDS with the 4-dword
    //      row pad applied by hardware (pad_interval=64 dwords, pad_amount=4).
    //      Issued once by wave 0; runs async while all waves fetch A-fragments.
    if (wave == 0) {
        unsigned long long ga = (unsigned long long)emb;
        unsigned lds_off = (unsigned)(unsigned long long)(&lds_emb[0]); // generic lo32 == LDS offset
        v4u g0;
        g0.x = 1u;                                          // count=1, is_restore=0, gather off
        g0.y = lds_off;                                     // [63:32]  lds_addr (bytes)
        g0.z = (unsigned)(ga & 0xFFFFFFFFu);                // [95:64]  global_addr lo
        g0.w = (unsigned)((ga >> 32) & 0x01FFFFFFu)         // [120:96] global_addr hi
             | (2u << 30);                                  // [127:126] type=2 (image)
        v8i g1;
        g1[0] = (2 << 16)      // data_size = 4 bytes
              | (1 << 20)      // pad_enable
              | (5 << 22)      // pad_interval: 2^(5+1)=64 dwords (one 64-float row)
              | (3 << 25);     // pad_amount: 3+1 = 4 dwords -> stride 68 floats
        g1[1] = (int)(64u << 16);                 // tensor_dim0=64  ([47:32] atomic addr = 0)
        g1[2] = (int)(512u << 16);                // tensor_dim0 hi=0, tensor_dim1=512 lo
        g1[3] = (int)(64u << 16);                 // tensor_dim1 hi=0, tile_dim0=64
        g1[4] = (int)512;                         // tile_dim1=512, tile_dim2=0
        g1[5] = 64;                               // tensor_dim0_stride = 64 elements
        g1[6] = 0;                                // stride hi, tensor_dim1_stride lo (unused, 2D)
        g1[7] = 0;
        v4i g2 = {0, 0, 0, 0};
        v4i g3 = {0, 0, 0, 0};
        v8i g4 = {0, 0, 0, 0, 0, 0, 0, 0};        // extra group (unused, 2D tensor)
        __builtin_amdgcn_tensor_load_to_lds(g0, g1, g2, g3, g4, 0);
    }

    // ---- load A fragments: 16 rows of x (NCHW gather, contiguous per half-wave);
    //      overlaps with the TDM transfer ----
    const int  rowbase  = (blockIdx.x * 8 + wave) * 16;     // multiple of 16 -> same (b,h)
    const int  n_m      = rowbase + m;                      // this lane's row
    const long lane_off = (long)(n_m >> 12) * 262144 + (n_m & 4095); // b*64*64*64 + h*64+w

    v2f a[16];
    float sq = 0.f;
    #pragma unroll
    for (int kk = 0; kk < 16; ++kk) {
        int c = 4 * kk + koff;
        float x0 = in[lane_off + (long)c * 4096];
        float x1 = in[lane_off + (long)(c + 1) * 4096];
        a[kk].x = x0; a[kk].y = x1;
        sq += x0 * x0 + x1 * x1;
    }
    sq += __shfl_xor(sq, 16);                 // lane & lane^16 together cover all 64 c's
    if (lane < 16) lds_xnorm[wave * 16 + lane] = sq;

    if (wave == 0) __builtin_amdgcn_s_wait_tensorcnt(0);   // TDM done before publishing LDS
    __syncthreads();

    // ---- ||e_k||^2 from the staged (padded) codebook ----
    {
        #pragma unroll
        for (int rr = 0; rr < 2; ++rr) {
            int r = tid * 2 + rr;
            float s = 0.f;
            #pragma unroll
            for (int c = 0; c < DIM; c += 4) {
                v4f v = *(const v4f*)(lds_emb + r * ESTR + c);
                s += v.x*v.x + v.y*v.y + v.z*v.z + v.w*v.w;
            }
            lds_enorm[r] = s;
        }
    }
    __syncthreads();

    float xn[8];
    #pragma unroll
    for (int j = 0; j < 8; ++j) xn[j] = lds_xnorm[wave * 16 + j + 8 * hi];

    float bestd[8]; int besti[8];
    #pragma unroll
    for (int j = 0; j < 8; ++j) { bestd[j] = 3.4e38f; besti[j] = 0; }

    float* dist = out + OFF_DIST;

    // ---- sweep all 32 column tiles: 4 independent WMMA chains (low occupancy
    //      from 139KB LDS -> in-wave ILP hides the matrix-pipe latency) ----
    for (int t = 0; t < 32; ++t) {
        int   col = t * 16 + m;
        float en  = lds_enorm[col];
        v8f acc0 = {0.f,0.f,0.f,0.f,0.f,0.f,0.f,0.f};
        v8f acc1 = acc0, acc2 = acc0, acc3 = acc0;
        #pragma unroll
        for (int kk = 0; kk < 16; kk += 4) {
            v2f b0 = *(const v2f*)(lds_emb + col * ESTR + 4 * (kk + 0) + koff);
            v2f b1 = *(const v2f*)(lds_emb + col * ESTR + 4 * (kk + 1) + koff);
            v2f b2 = *(const v2f*)(lds_emb + col * ESTR + 4 * (kk + 2) + koff);
            v2f b3 = *(const v2f*)(lds_emb + col * ESTR + 4 * (kk + 3) + koff);
            acc0 = __builtin_amdgcn_wmma_f32_16x16x4_f32(false, a[kk+0], false, b0,
                                                         (short)0, acc0, false, false);
            acc1 = __builtin_amdgcn_wmma_f32_16x16x4_f32(false, a[kk+1], false, b1,
                                                         (short)0, acc1, false, false);
            acc2 = __builtin_amdgcn_wmma_f32_16x16x4_f32(false, a[kk+2], false, b2,
                                                         (short)0, acc2, false, false);
            acc3 = __builtin_amdgcn_wmma_f32_16x16x4_f32(false, a[kk+3], false, b3,
                                                         (short)0, acc3, false, false);
        }
        #pragma unroll
        for (int j = 0; j < 8; ++j) {
            float dot = (acc0[j] + acc1[j]) + (acc2[j] + acc3[j]);
            float d = xn[j] + en - 2.0f * dot;
            dist[(long)(rowbase + j + 8 * hi) * NEMB + col] = d;
            if (d < bestd[j]) { bestd[j] = d; besti[j] = col; }  // cols ascend -> first-min kept
        }
    }

    // ---- argmin butterfly across the 16 lanes sharing each row ----
    #pragma unroll
    for (int j = 0; j < 8; ++j) {
        #pragma unroll
        for (int mask = 1; mask <= 8; mask <<= 1) {
            float od = __shfl_xor(bestd[j], mask);
            int   oi = __shfl_xor(besti[j], mask);
            if (od < bestd[j] || (od == bestd[j] && oi < besti[j])) { bestd[j] = od; besti[j] = oi; }
        }
        if (m == j) {   // one lane per row emits index, histogram, LDS stash
            int r = rowbase + j + 8 * hi;
            out[OFF_IDX + r] = (float)besti[j];
            atomicAdd(&hist[besti[j]], 1u);
            lds_argidx[wave * 16 + j + 8 * hi] = besti[j];
        }
    }
    // same-wave LDS RAW: DS ops are in-order per wave, no barrier needed

    // ---- one-hot encodings + NHWC quantized (coalesced b128/b64 stores) ----
    float* enc   = out + OFF_ENC;
    float* qloss = out + OFF_QLOSS;
    for (int mm = 0; mm < 16; ++mm) {
        int  k_m = lds_argidx[wave * 16 + mm];   // uniform broadcast read
        long r   = rowbase + mm;
        #pragma unroll
        for (int i = 0; i < 4; ++i) {
            int cb = lane * 4 + i * 128;
            v4f v;
            v.x = (cb + 0 == k_m) ? 1.f : 0.f;
            v.y = (cb + 1 == k_m) ? 1.f : 0.f;
            v.z = (cb + 2 == k_m) ? 1.f : 0.f;
            v.w = (cb + 3 == k_m) ? 1.f : 0.f;
            *(v4f*)(enc + r * NEMB + cb) = v;
        }
        v2f q = *(const v2f*)(lds_emb + k_m * ESTR + lane * 2);
        *(v2f*)(qloss + r * DIM + lane * 2) = q;
    }

    // ---- NCHW quantized_out (straight-through == quantized in forward) ----
    float* qout = out + OFF_QOUT;
    int kreg = lds_argidx[wave * 16 + m];
    #pragma unroll
    for (int cc = 0; cc < 32; ++cc) {
        int c = cc * 2 + hi;                       // halves interleave c; 64B/half-wave stores
        qout[lane_off + (long)c * 4096] = lds_emb[kreg * ESTR + c];
    }
}

__global__ __launch_bounds__(512) void vq_perplexity(const unsigned int* __restrict__ hist,
                                                     float* __restrict__ out)
{
    __shared__ float red[512];
    int t = threadIdx.x;
    float p = (float)hist[t] * (1.0f / (float)NTOT);
    red[t] = p * logf(p + 1e-10f);
    __syncthreads();
    for (int s = 256; s > 0; s >>= 1) {
        if (t < s) red[t] += red[t + s];
        __syncthreads();
    }
    if (t == 0) out[OFF_PERP] = expf(-red[0]);
}

extern "C" void kernel_launch(void* const* d_in, const int* in_sizes, int n_in,
                              void* d_out, int out_size, void* d_ws, size_t ws_size,
                              hipStream_t stream) {
    (void)in_sizes; (void)n_in; (void)out_size; (void)ws_size;
    const float* in  = (const float*)d_in[0];
    const float* emb = (const float*)d_in[1];
    float* out = (float*)d_out;
    unsigned int* hist = (unsigned int*)d_ws;

    vq_zero_hist<<<1, 512, 0, stream>>>(hist);
    vq_main<<<1024, 256, 0, stream>>>(in, emb, out, hist);
    vq_perplexity<<<1, 512, 0, stream>>>(hist, out);
}